// Downsample_19215683682639
// MI455X (gfx1250) — compile-verified
//
#include <hip/hip_runtime.h>
#include <hip/hip_bf16.h>

typedef __attribute__((ext_vector_type(2))) float v2f;
typedef __attribute__((ext_vector_type(8))) float v8f;
typedef __attribute__((ext_vector_type(4))) unsigned int v4u;
typedef __attribute__((ext_vector_type(4))) int v4i;
typedef __attribute__((ext_vector_type(8))) int v8i;

#define BB 8
#define CC 128
#define NN 4096
#define MM 1024
#define KNN 32
#define LREL 0.2f
#define BNEPS 1e-5f
#define WS 130   // padded LDS row stride (floats) for weight tiles: 128 + 2 (TDM pad)

// Detect therock/amdgpu-toolchain (6-arg tensor_load_to_lds builtin); the TDM
// header ships only with that toolchain.
#if defined(__has_include)
#  if __has_include(<hip/amd_detail/amd_gfx1250_TDM.h>)
#    define ATH_TDM_6ARG 1
#  endif
#endif

#if defined(__HIP_DEVICE_COMPILE__) && __has_builtin(__builtin_amdgcn_tensor_load_to_lds)
#define ATH_HAVE_TDM 1
// Issue one TDM DMA of a 2D tile (tileD1 rows x tileD0 4-byte elements) from a
// row-major tensor (row stride strideD0 elements) into LDS at byte offset
// ldsOff, padding the LDS destination by 2 dwords every 128 dwords
// (-> 130-float LDS row stride, bank-conflict-free A-fragment reads).
// Descriptor bitfields per CDNA5 ISA.
__device__ __forceinline__ void tdm_load_2d_padded(unsigned ldsOff, const void* gptr,
                                                   unsigned tensorD0, unsigned tensorD1,
                                                   unsigned strideD0,
                                                   unsigned tileD0, unsigned tileD1) {
  const unsigned long long ga = (unsigned long long)gptr;
  v4u g0;
  g0[0] = 1u;                                                   // count=1, user D#
  g0[1] = ldsOff;                                               // lds_addr
  g0[2] = (unsigned)(ga & 0xFFFFFFFFu);                         // global_addr[31:0]
  g0[3] = (unsigned)((ga >> 32) & 0x01FFFFFFu) | 0x80000000u;   // addr[56:32] | type=2
  v8i g1;
  g1[0] = (int)((2u << 16)      // data_size = 4 bytes
              | (1u << 20)      // pad_enable
              | (6u << 22)      // pad_interval: every 128 dwords
              | (1u << 25));    // pad_amount: 2 dwords
  g1[1] = (int)((tensorD0 & 0xFFFFu) << 16);                    // tensor_dim0[15:0]
  g1[2] = (int)(((tensorD0 >> 16) & 0xFFFFu) | ((tensorD1 & 0xFFFFu) << 16));
  g1[3] = (int)(((tensorD1 >> 16) & 0xFFFFu) | ((tileD0 & 0xFFFFu) << 16));
  g1[4] = (int)(tileD1 & 0xFFFFu);                              // tile_dim1 (tile_dim2=0)
  g1[5] = (int)strideD0;                                        // tensor_dim0_stride lo
  g1[6] = 0;
  g1[7] = 0;
  v4i z4 = {0, 0, 0, 0};
#ifdef ATH_TDM_6ARG
  v8i z8 = {0, 0, 0, 0, 0, 0, 0, 0};
  __builtin_amdgcn_tensor_load_to_lds(g0, g1, z4, z4, z8, 0);
#else
  __builtin_amdgcn_tensor_load_to_lds(g0, g1, z4, z4, 0);
#endif
}
#endif

// ---------------------------------------------------------------------------
// Prep: per-point squared norms of x  (xnorm[b][n])
// ---------------------------------------------------------------------------
__global__ void prep_xnorm_kernel(const float* __restrict__ x, float* __restrict__ xnorm) {
  const int t = blockIdx.x * 256 + threadIdx.x;      // BB*NN threads
  const int b = t / NN, n = t % NN;
  float s = 0.f;
  for (int c = 0; c < CC; ++c) {
    const float v = x[((size_t)b * CC + c) * NN + n];
    s += v * v;
  }
  xnorm[t] = s;
}

// Wc[o][i] = W1[o][128+i] - W1[o][i]   (center-term weights after fusion)
__global__ void prep_wc_kernel(const float* __restrict__ W1, float* __restrict__ Wc) {
  const int e = blockIdx.x * 256 + threadIdx.x;      // 128*128 threads
  const int o = e >> 7, i = e & 127;
  Wc[e] = W1[(size_t)o * 256 + 128 + i] - W1[(size_t)o * 256 + i];
}

// ---------------------------------------------------------------------------
// Farthest point sampling: one workgroup per batch, 1024 threads, 4 pts/thread
// ---------------------------------------------------------------------------
__global__ void __launch_bounds__(1024) fps_kernel(const float* __restrict__ x,
                                                   int* __restrict__ idx) {
  const int b = blockIdx.x;
  const int tid = threadIdx.x;
  const int lane = tid & 31, wave = tid >> 5;
  __shared__ float sLast[CC];
  __shared__ float sVal[32];
  __shared__ int   sIdxW[32];
  __shared__ int   sBestN;
  const float* xb = x + (size_t)b * CC * NN;

  if (tid < CC) sLast[tid] = xb[(size_t)tid * NN];   // point 0 coords
  if (tid == 0) idx[b * MM] = 0;
  float4 dist = make_float4(1e10f, 1e10f, 1e10f, 1e10f);
  __syncthreads();

  for (int s = 1; s < MM; ++s) {
    float4 dd = make_float4(0.f, 0.f, 0.f, 0.f);
#pragma unroll 4
    for (int c = 0; c < CC; ++c) {
      const float lp = sLast[c];
      const float4 v = reinterpret_cast<const float4*>(xb + (size_t)c * NN)[tid];
      const float tx = v.x - lp, ty = v.y - lp, tz = v.z - lp, tw = v.w - lp;
      dd.x += tx * tx; dd.y += ty * ty; dd.z += tz * tz; dd.w += tw * tw;
    }
    dist.x = fminf(dist.x, dd.x); dist.y = fminf(dist.y, dd.y);
    dist.z = fminf(dist.z, dd.z); dist.w = fminf(dist.w, dd.w);
    // local argmax (first-max wins like jnp.argmax)
    float bv = dist.x; int bn = tid * 4;
    if (dist.y > bv) { bv = dist.y; bn = tid * 4 + 1; }
    if (dist.z > bv) { bv = dist.z; bn = tid * 4 + 2; }
    if (dist.w > bv) { bv = dist.w; bn = tid * 4 + 3; }
    for (int off = 16; off; off >>= 1) {
      const float ov = __shfl_xor(bv, off, 32);
      const int   on = __shfl_xor(bn, off, 32);
      if (ov > bv || (ov == bv && on < bn)) { bv = ov; bn = on; }
    }
    if (lane == 0) { sVal[wave] = bv; sIdxW[wave] = bn; }
    __syncthreads();
    if (wave == 0) {
      float v2 = sVal[lane]; int n2 = sIdxW[lane];
      for (int off = 16; off; off >>= 1) {
        const float ov = __shfl_xor(v2, off, 32);
        const int   on = __shfl_xor(n2, off, 32);
        if (ov > v2 || (ov == v2 && on < n2)) { v2 = ov; n2 = on; }
      }
      if (lane == 0) { sBestN = n2; idx[b * MM + s] = n2; }
    }
    __syncthreads();
    const int bestn = sBestN;
    if (tid < CC) sLast[tid] = xb[(size_t)tid * NN + bestn];
    __syncthreads();
  }
}

// ---------------------------------------------------------------------------
// Gather sampled points x_ds[b][c][m] and their squared norms
// ---------------------------------------------------------------------------
__global__ void gather_kernel(const float* __restrict__ x, const int* __restrict__ idx,
                              float* __restrict__ xds, float* __restrict__ qnorm) {
  const int t = blockIdx.x * 256 + threadIdx.x;      // BB*MM threads
  const int b = t / MM, m = t % MM;
  const int n = idx[t];
  float s = 0.f;
  for (int c = 0; c < CC; ++c) {
    const float v = x[((size_t)b * CC + c) * NN + n];
    xds[((size_t)b * CC + c) * MM + m] = v;
    s += v * v;
  }
  qnorm[t] = s;
}

// ---------------------------------------------------------------------------
// kNN: per block, 16 query rows x full N=4096 distance tile in LDS (256KB),
// distance dot-products via v_wmma_f32_16x16x4_f32, then 32x wave argmin.
// grid = BB*64, block = 512 (16 waves)
// ---------------------------------------------------------------------------
__global__ void __launch_bounds__(512) knn_kernel(const float* __restrict__ xds,
                                                  const float* __restrict__ x,
                                                  const float* __restrict__ qnorm,
                                                  const float* __restrict__ xnorm,
                                                  int* __restrict__ knn) {
  extern __shared__ float sD[];                       // 16 * 4096 floats
  const int bx = blockIdx.x;
  const int b = bx >> 6, mt = bx & 63, m0 = mt * 16;
  const int tid = threadIdx.x, lane = tid & 31, wave = tid >> 5;
  const float* qb = xds + (size_t)b * CC * MM;
  const float* xb = x + (size_t)b * CC * NN;

  for (int i = 0; i < 16; ++i) {
    const int nt = wave + 16 * i, n0 = nt * 16;
    const int row  = m0 + (lane & 15);
    const int col  = n0 + (lane & 15);
    const int koff = 2 * (lane >> 4);
    v8f acc = {};
#pragma unroll 4
    for (int kk = 0; kk < CC; kk += 4) {
      v2f a, bf;
      a.x  = qb[(size_t)(kk + koff) * MM + row];
      a.y  = qb[(size_t)(kk + koff + 1) * MM + row];
      bf.x = xb[(size_t)(kk + koff) * NN + col];
      bf.y = xb[(size_t)(kk + koff + 1) * NN + col];
      acc = __builtin_amdgcn_wmma_f32_16x16x4_f32(false, a, false, bf,
                                                  (short)0, acc, false, false);
    }
    const float xn = xnorm[b * NN + col];
#pragma unroll
    for (int r = 0; r < 8; ++r) {
      const int mrow = r + 8 * (lane >> 4);
      sD[mrow * NN + col] = qnorm[b * MM + m0 + mrow] + xn - 2.0f * acc[r];
    }
  }
  __syncthreads();

  // top-32 smallest per row; wave w owns row w
  float* rowD = sD + wave * NN;
  int* kout = knn + ((size_t)b * MM + m0 + wave) * KNN;
  for (int r = 0; r < KNN; ++r) {
    float bv = 3.0e38f; int bn = NN;
    for (int i = 0; i < NN / 32; ++i) {
      const int n = lane + i * 32;
      const float v = rowD[n];
      if (v < bv || (v == bv && n < bn)) { bv = v; bn = n; }
    }
    for (int off = 16; off; off >>= 1) {
      const float ov = __shfl_xor(bv, off, 32);
      const int   on = __shfl_xor(bn, off, 32);
      if (ov < bv || (ov == bv && on < bn)) { bv = ov; bn = on; }
    }
    if (lane == 0) { kout[r] = bn; rowD[bn] = 3.0e38f; }
    __syncthreads();
  }
}

// ---------------------------------------------------------------------------
// Fused conv tile: per block handles (b, 4 m's) -> S=128 columns.
//   Weights W1a/W2 staged to LDS via tensor_load_to_lds (TDM), overlapped
//   with the irregular neighbor gather; waited with s_wait_tensorcnt.
//   h1 = W1a @ gathered_x + (W1b - W1a) @ center      (WMMA)
//   STAGE 0: emit per-channel sum/sumsq partials of h1
//   STAGE 1: BN1+lrelu, h2 = W2 @ h1n (WMMA), emit partials of h2
//   STAGE 2: full chain, BN2+lrelu, max over k -> out
// grid = BB*256, block = 512 (16 waves)
// ---------------------------------------------------------------------------
template <int STAGE>
__global__ void __launch_bounds__(512) conv_pass(
    const float* __restrict__ x, const float* __restrict__ xds,
    const int* __restrict__ knn, const float* __restrict__ W1,
    const float* __restrict__ Wc, const float* __restrict__ W2,
    const float* __restrict__ g1, const float* __restrict__ b1,
    const float* __restrict__ g2, const float* __restrict__ b2,
    const float* __restrict__ stats1, const float* __restrict__ stats2,
    float* __restrict__ part, float* __restrict__ out) {
  extern __shared__ char smem[];
  float* sNeigh = (float*)smem;            // 128 x 128
  float* sH     = sNeigh + 128 * 128;      // 128 x 128
  float* sQ     = sH + 128 * 128;          // 128 x 4
  float* sCb    = sQ + 512;                // 128 x 4
  float* sRed   = sCb + 512;               // 4 x 128 x 2
  float* sW1    = sRed + 1024;             // 128 x 130 (padded W1a tile)
  float* sW2    = sW1 + 128 * WS;          // 128 x 130 (padded W2 tile)
  int*   sIdx   = (int*)(sW2 + 128 * WS);  // 128

  const int bx = blockIdx.x;
  const int b = bx >> 8, mt = bx & 255, m0 = mt * 4;
  const int tid = threadIdx.x, lane = tid & 31, wave = tid >> 5;
  const int S = 128;
  const float* xb = x + (size_t)b * CC * NN;

#if defined(ATH_HAVE_TDM)
  if (tid < 32) {   // wave 0 issues both weight-tile DMAs (EXEC-independent)
    tdm_load_2d_padded((unsigned)(unsigned long long)(void*)sW1, W1,
                       /*tensorD0=*/256, /*tensorD1=*/128, /*strideD0=*/256,
                       /*tileD0=*/128, /*tileD1=*/128);
    tdm_load_2d_padded((unsigned)(unsigned long long)(void*)sW2, W2,
                       /*tensorD0=*/128, /*tensorD1=*/128, /*strideD0=*/128,
                       /*tileD0=*/128, /*tileD1=*/128);
  }
#endif

  if (tid < 128)
    sIdx[tid] = knn[((size_t)b * MM + m0 + (tid >> 5)) * KNN + (tid & 31)];
  {
    const int c = tid >> 2, m = tid & 3;
    sQ[c * 4 + m] = xds[((size_t)b * CC + c) * MM + m0 + m];
  }
  __syncthreads();

  // gather neighbor tile [128 ch x 128 cols] (overlaps the TDM weight DMA)
  for (int j = 0; j < 32; ++j) {
    const int e = tid + j * 512;
    const int c = e >> 7, s = e & 127;
    sNeigh[c * S + s] = xb[(size_t)c * NN + sIdx[s]];
  }
  // center term cb[o][m]
  {
    const int o = tid >> 2, m = tid & 3;
    float acc = 0.f;
    for (int i = 0; i < 128; ++i) acc += Wc[o * 128 + i] * sQ[i * 4 + m];
    sCb[o * 4 + m] = acc;
  }
#if defined(ATH_HAVE_TDM)
  if (tid < 32) __builtin_amdgcn_s_wait_tensorcnt(0);
#else
  for (int j = tid; j < 128 * 128; j += 512) {   // fallback cooperative copy
    const int rr = j >> 7, cc2 = j & 127;
    sW1[rr * WS + cc2] = W1[(size_t)rr * 256 + cc2];
    sW2[rr * WS + cc2] = W2[(size_t)rr * 128 + cc2];
  }
#endif
  __syncthreads();

  // GEMM1: h1[o][s]
  for (int i = 0; i < 4; ++i) {
    const int tt = wave + 16 * i;
    const int o0 = (tt >> 3) * 16, s0 = (tt & 7) * 16;
    const int row = o0 + (lane & 15), col = s0 + (lane & 15);
    const int koff = 2 * (lane >> 4);
    v8f acc = {};
#pragma unroll 4
    for (int kk = 0; kk < 128; kk += 4) {
      v2f a, bf;
      a.x  = sW1[row * WS + kk + koff];
      a.y  = sW1[row * WS + kk + koff + 1];
      bf.x = sNeigh[(kk + koff) * S + col];
      bf.y = sNeigh[(kk + koff + 1) * S + col];
      acc = __builtin_amdgcn_wmma_f32_16x16x4_f32(false, a, false, bf,
                                                  (short)0, acc, false, false);
    }
#pragma unroll
    for (int r = 0; r < 8; ++r) {
      const int o = o0 + r + 8 * (lane >> 4);
      sH[o * S + col] = acc[r] + sCb[o * 4 + (col >> 5)];
    }
  }
  __syncthreads();

  if (STAGE == 0) {
    const int o = tid & 127, grp = tid >> 7;
    float s = 0.f, q = 0.f;
    for (int k = 0; k < 32; ++k) {
      const float v = sH[o * S + grp * 32 + k];
      s += v; q += v * v;
    }
    sRed[(grp * 128 + o) * 2] = s; sRed[(grp * 128 + o) * 2 + 1] = q;
    __syncthreads();
    if (tid < 128) {
      float ss = 0.f, qq = 0.f;
      for (int g = 0; g < 4; ++g) {
        ss += sRed[(g * 128 + tid) * 2];
        qq += sRed[(g * 128 + tid) * 2 + 1];
      }
      part[((size_t)bx * 128 + tid) * 2] = ss;
      part[((size_t)bx * 128 + tid) * 2 + 1] = qq;
    }
    return;
  }

  // BN1 + leaky-relu in place
  for (int j = tid; j < 128 * S; j += 512) {
    const int o = j >> 7;
    float h = (sH[j] - stats1[o * 2]) * stats1[o * 2 + 1] * g1[o] + b1[o];
    sH[j] = h > 0.f ? h : LREL * h;
  }
  __syncthreads();

  // GEMM2: h2 = W2 @ h1n, into sH2 (reuse sNeigh)
  float* sH2 = sNeigh;
  for (int i = 0; i < 4; ++i) {
    const int tt = wave + 16 * i;
    const int o0 = (tt >> 3) * 16, s0 = (tt & 7) * 16;
    const int row = o0 + (lane & 15), col = s0 + (lane & 15);
    const int koff = 2 * (lane >> 4);
    v8f acc = {};
#pragma unroll 4
    for (int kk = 0; kk < 128; kk += 4) {
      v2f a, bf;
      a.x  = sW2[row * WS + kk + koff];
      a.y  = sW2[row * WS + kk + koff + 1];
      bf.x = sH[(kk + koff) * S + col];
      bf.y = sH[(kk + koff + 1) * S + col];
      acc = __builtin_amdgcn_wmma_f32_16x16x4_f32(false, a, false, bf,
                                                  (short)0, acc, false, false);
    }
#pragma unroll
    for (int r = 0; r < 8; ++r) {
      const int o = o0 + r + 8 * (lane >> 4);
      sH2[o * S + col] = acc[r];
    }
  }
  __syncthreads();

  if (STAGE == 1) {
    const int o = tid & 127, grp = tid >> 7;
    float s = 0.f, q = 0.f;
    for (int k = 0; k < 32; ++k) {
      const float v = sH2[o * S + grp * 32 + k];
      s += v; q += v * v;
    }
    sRed[(grp * 128 + o) * 2] = s; sRed[(grp * 128 + o) * 2 + 1] = q;
    __syncthreads();
    if (tid < 128) {
      float ss = 0.f, qq = 0.f;
      for (int g = 0; g < 4; ++g) {
        ss += sRed[(g * 128 + tid) * 2];
        qq += sRed[(g * 128 + tid) * 2 + 1];
      }
      part[((size_t)bx * 128 + tid) * 2] = ss;
      part[((size_t)bx * 128 + tid) * 2 + 1] = qq;
    }
    return;
  }

  // STAGE 2: BN2 + leaky-relu, then max over k neighbors
  for (int j = tid; j < 128 * S; j += 512) {
    const int o = j >> 7;
    float h = (sH2[j] - stats2[o * 2]) * stats2[o * 2 + 1] * g2[o] + b2[o];
    sH2[j] = h > 0.f ? h : LREL * h;
  }
  __syncthreads();
  {
    const int o = tid >> 2, m = tid & 3;
    float mx = -3.0e38f;
    for (int k = 0; k < KNN; ++k) mx = fmaxf(mx, sH2[o * S + m * 32 + k]);
    out[((size_t)b * 128 + o) * MM + m0 + m] = mx;
  }
}

// ---------------------------------------------------------------------------
// Deterministic reduction of per-workgroup BN partials -> (mean, rstd)
// ---------------------------------------------------------------------------
__global__ void reduce_stats_kernel(const float* __restrict__ part,
                                    float* __restrict__ stats, int nwg, float count) {
  const int o = threadIdx.x;   // 128 threads
  float s = 0.f, q = 0.f;
  for (int w = 0; w < nwg; ++w) {
    s += part[((size_t)w * 128 + o) * 2];
    q += part[((size_t)w * 128 + o) * 2 + 1];
  }
  const float mean = s / count;
  const float var = q / count - mean * mean;
  stats[o * 2] = mean;
  stats[o * 2 + 1] = rsqrtf(var + BNEPS);
}

// FPS indices appended to output as floats
__global__ void write_idx_kernel(const int* __restrict__ idx, float* __restrict__ out) {
  const int j = blockIdx.x * 256 + threadIdx.x;   // BB*MM threads
  out[(size_t)BB * 128 * MM + j] = (float)idx[j];
}

// ---------------------------------------------------------------------------
extern "C" void kernel_launch(void* const* d_in, const int* in_sizes, int n_in,
                              void* d_out, int out_size, void* d_ws, size_t ws_size,
                              hipStream_t stream) {
  const float* x  = (const float*)d_in[0];
  const float* W1 = (const float*)d_in[1];
  const float* g1 = (const float*)d_in[2];
  const float* b1 = (const float*)d_in[3];
  const float* W2 = (const float*)d_in[4];
  const float* g2 = (const float*)d_in[5];
  const float* b2 = (const float*)d_in[6];
  float* out = (float*)d_out;

  char* w = (char*)d_ws;
  size_t off = 0;
  auto take = [&](size_t bytes) -> void* {
    void* p = w + off;
    off = (off + bytes + 255) & ~(size_t)255;
    return p;
  };
  int*   idxBuf = (int*)take((size_t)BB * MM * 4);
  float* xds    = (float*)take((size_t)BB * CC * MM * 4);
  float* qnorm  = (float*)take((size_t)BB * MM * 4);
  float* xnorm  = (float*)take((size_t)BB * NN * 4);
  float* Wc     = (float*)take((size_t)128 * 128 * 4);
  int*   knnIdx = (int*)take((size_t)BB * MM * KNN * 4);
  float* part1  = (float*)take((size_t)BB * 256 * 128 * 2 * 4);
  float* part2  = (float*)take((size_t)BB * 256 * 128 * 2 * 4);
  float* stats1 = (float*)take(256 * 4);
  float* stats2 = (float*)take(256 * 4);

  const int nwgConv = BB * 256;                       // 2048
  const float cnt = (float)((size_t)BB * MM * KNN);   // 262144 per channel
  const size_t knnLds  = (size_t)16 * NN * 4;         // 256 KB
  const size_t convLds =
      (size_t)(2 * 128 * 128 + 512 + 512 + 1024 + 2 * 128 * WS) * 4 + 512; // ~267 KB

  prep_xnorm_kernel<<<(BB * NN) / 256, 256, 0, stream>>>(x, xnorm);
  prep_wc_kernel<<<(128 * 128) / 256, 256, 0, stream>>>(W1, Wc);
  fps_kernel<<<BB, 1024, 0, stream>>>(x, idxBuf);
  gather_kernel<<<(BB * MM) / 256, 256, 0, stream>>>(x, idxBuf, xds, qnorm);
  knn_kernel<<<BB * 64, 512, knnLds, stream>>>(xds, x, qnorm, xnorm, knnIdx);

  conv_pass<0><<<nwgConv, 512, convLds, stream>>>(x, xds, knnIdx, W1, Wc, W2,
      g1, b1, g2, b2, stats1, stats2, part1, out);
  reduce_stats_kernel<<<1, 128, 0, stream>>>(part1, stats1, nwgConv, cnt);
  conv_pass<1><<<nwgConv, 512, convLds, stream>>>(x, xds, knnIdx, W1, Wc, W2,
      g1, b1, g2, b2, stats1, stats2, part2, out);
  reduce_stats_kernel<<<1, 128, 0, stream>>>(part2, stats2, nwgConv, cnt);
  conv_pass<2><<<nwgConv, 512, convLds, stream>>>(x, xds, knnIdx, W1, Wc, W2,
      g1, b1, g2, b2, stats1, stats2, part2, out);

  write_idx_kernel<<<(BB * MM) / 256, 256, 0, stream>>>(idxBuf, out);
}